// LimnetLayer_42838003810566
// MI455X (gfx1250) — compile-verified
//
#include <hip/hip_runtime.h>
#include <math.h>

#define B_SZ      1024
#define N_USERS   1000
#define E         128
#define UF        32
#define IN_COLS   66           // 2 + UF + IF
#define K_DIM     320
#define N_DIM     384
#define OUT_COLS  258
#define MEM_ELEMS (B_SZ * N_USERS * E)   // 131,072,000 floats per memory tensor
#define A_PITCH   324          // 320 + 4 pad: breaks 64-bank aliasing (320 % 64 == 0)

typedef float v2f __attribute__((ext_vector_type(2)));
typedef float v8f __attribute__((ext_vector_type(8)));

// ---------------------------------------------------------------------------
// Kernel 1: bulk copy of user_memory / item_memory into d_out (bandwidth bound,
// ~2.1 GB of traffic -> this dominates total runtime). float4 grid-stride.
// ---------------------------------------------------------------------------
__global__ void limnet_copy_mem(const float4* __restrict__ um,
                                const float4* __restrict__ im,
                                float4* __restrict__ out_um,
                                float4* __restrict__ out_im) {
    const long long n4 = MEM_ELEMS / 4;
    const long long stride = (long long)gridDim.x * blockDim.x;
    for (long long i = (long long)blockIdx.x * blockDim.x + threadIdx.x; i < n4; i += stride) {
        out_um[i] = um[i];
        out_im[i] = im[i];
    }
}

// ---------------------------------------------------------------------------
// Kernel 2: fused gather -> f32 WMMA GEMM (16x16x4) -> GRU -> L2 norm -> scatter.
// grid.x = 64 row-blocks of 16, grid.y = {0:user GEMM, 1:item GEMM}.
// 768 threads = 24 waves; wave w owns output-tile columns [16w, 16w+16).
// ---------------------------------------------------------------------------
__global__ __launch_bounds__(768)
void limnet_fused(const float* __restrict__ inputs,
                  const float* __restrict__ user_mem,
                  const float* __restrict__ item_mem,
                  const float* __restrict__ w_u, const float* __restrict__ bih_u, const float* __restrict__ bhh_u,
                  const float* __restrict__ w_i, const float* __restrict__ bih_i, const float* __restrict__ bhh_i,
                  float* __restrict__ d_out) {
    __shared__ float xlds[16][A_PITCH];     // 16 x 320 input tile (padded pitch)
    __shared__ float gilds[16][N_DIM];      // 16 x 384 GEMM result
    __shared__ float nt[16][E];             // GRU output before normalization
    __shared__ float invnorm[16];
    __shared__ int   uid_s[16];
    __shared__ int   iid_s[16];

    const int tid  = threadIdx.x;
    const int g    = blockIdx.y;            // 0 = user path, 1 = item path
    const int row0 = blockIdx.x * 16;

    const float* w   = g ? w_i   : w_u;
    const float* bih = g ? bih_i : bih_u;
    const float* bhh = g ? bhh_i : bhh_u;

    // ---- phase 0: read ids for the 16 rows; emit id columns of `out` once ----
    if (tid < 16) {
        const int b  = row0 + tid;
        const int u  = (int)inputs[b * IN_COLS + 0];
        const int it = (int)inputs[b * IN_COLS + 1];
        uid_s[tid] = u;
        iid_s[tid] = it;
        if (g == 0) {
            d_out[(long long)b * OUT_COLS + 0] = (float)u;
            d_out[(long long)b * OUT_COLS + 1] = (float)it;
        }
    }
    __syncthreads();

    // ---- phase 1: gather A tile into LDS.
    // item_inputs[b][k] == user_inputs[b][(k+160) % 320] (halves swapped).
    for (int idx = tid; idx < 16 * K_DIM; idx += 768) {
        const int r  = idx / K_DIM;
        const int k  = idx - r * K_DIM;
        const int b  = row0 + r;
        const int kk = g ? ((k + 160) % K_DIM) : k;   // map to user-layout column
        float v;
        if (kk < E) {
            v = user_mem[((long long)b * N_USERS + uid_s[r]) * E + kk];
        } else if (kk < E + UF) {
            v = inputs[b * IN_COLS + 2 + (kk - E)];
        } else if (kk < 2 * E + UF) {
            v = item_mem[((long long)b * N_USERS + iid_s[r]) * E + (kk - E - UF)];
        } else {
            v = inputs[b * IN_COLS + 2 + UF + (kk - 2 * E - UF)];
        }
        xlds[r][k] = v;
    }
    __syncthreads();

    // ---- phase 2: WMMA GEMM. gi[m][n] = sum_k x[m][k] * W[n][k]  (+ b_ih[n]).
    // f32 16x16x4 layout: lane l: idx = l&15, half = l>>4; operand VGPR j holds
    // K = 4*iter + 2*half + j. A indexed by M=idx, B by N=idx (B = W^T).
    const int wid  = tid >> 5;
    const int lane = tid & 31;
    const int half = lane >> 4;
    const int l15  = lane & 15;
    const int n0   = wid * 16;

    v8f acc;
#pragma unroll
    for (int p = 0; p < 8; ++p) acc[p] = 0.0f;

    const float* wrow = w + (long long)(n0 + l15) * K_DIM;
    for (int k4 = 0; k4 < K_DIM; k4 += 4) {
        const int kbase = k4 + 2 * half;
        v2f a, bf;
        a.x  = xlds[l15][kbase];
        a.y  = xlds[l15][kbase + 1];
        bf.x = wrow[kbase];
        bf.y = wrow[kbase + 1];
        acc = __builtin_amdgcn_wmma_f32_16x16x4_f32(false, a, false, bf,
                                                    (short)0, acc, false, false);
    }

    // D layout: lane l, VGPR p -> row = p + 8*half, col = l&15.
    const float biasv = bih[n0 + l15];
#pragma unroll
    for (int p = 0; p < 8; ++p) {
        gilds[p + 8 * half][n0 + l15] = acc[p] + biasv;
    }
    __syncthreads();

    // ---- phase 3: GRU with zero hidden state ----
    for (int idx = tid; idx < 16 * E; idx += 768) {
        const int r = idx >> 7;
        const int c = idx & (E - 1);
        const float rg = gilds[r][c]         + bhh[c];
        const float zg = gilds[r][E + c]     + bhh[E + c];
        const float rv = 1.0f / (1.0f + __expf(-rg));
        const float zv = 1.0f / (1.0f + __expf(-zg));
        const float nv = tanhf(gilds[r][2 * E + c] + rv * bhh[2 * E + c]);
        nt[r][c] = (1.0f - zv) * nv;
    }
    __syncthreads();

    // ---- phase 4: L2 norm per row (wave32 shuffle reduction, one wave/row) ----
    if (wid < 16) {
        const int r = wid;
        float s = 0.0f;
        for (int c = lane; c < E; c += 32) {
            const float v = nt[r][c];
            s += v * v;
        }
        for (int off = 16; off > 0; off >>= 1) s += __shfl_xor(s, off, 32);
        if (lane == 0) invnorm[r] = 1.0f / fmaxf(sqrtf(s), 1e-12f);
    }
    __syncthreads();

    // ---- phase 5: write packed output + scatter updated row into memory copy ----
    float* mem_out = d_out + (long long)B_SZ * OUT_COLS + (long long)g * MEM_ELEMS;
    for (int idx = tid; idx < 16 * E; idx += 768) {
        const int r = idx >> 7;
        const int c = idx & (E - 1);
        const int b = row0 + r;
        const float v = nt[r][c] * invnorm[r];
        d_out[(long long)b * OUT_COLS + 2 + g * E + c] = v;
        const int id = g ? iid_s[r] : uid_s[r];
        mem_out[((long long)b * N_USERS + id) * E + c] = v;
    }
}

extern "C" void kernel_launch(void* const* d_in, const int* in_sizes, int n_in,
                              void* d_out, int out_size, void* d_ws, size_t ws_size,
                              hipStream_t stream) {
    (void)in_sizes; (void)n_in; (void)out_size; (void)d_ws; (void)ws_size;

    const float* inputs = (const float*)d_in[0];
    const float* um     = (const float*)d_in[1];
    const float* im     = (const float*)d_in[2];
    const float* w_u    = (const float*)d_in[3];
    const float* bih_u  = (const float*)d_in[4];
    const float* bhh_u  = (const float*)d_in[5];
    const float* w_i    = (const float*)d_in[6];
    const float* bih_i  = (const float*)d_in[7];
    const float* bhh_i  = (const float*)d_in[8];

    float* out    = (float*)d_out;
    float* out_um = out + (size_t)B_SZ * OUT_COLS;
    float* out_im = out_um + (size_t)MEM_ELEMS;

    // 1) bulk memory passthrough (dominant cost, pure bandwidth)
    limnet_copy_mem<<<8192, 256, 0, stream>>>((const float4*)um, (const float4*)im,
                                              (float4*)out_um, (float4*)out_im);

    // 2) fused gather + WMMA GEMM + GRU + normalize + scatter (after copy, same stream)
    dim3 grid(B_SZ / 16, 2);
    limnet_fused<<<grid, 768, 0, stream>>>(inputs, um, im,
                                           w_u, bih_u, bhh_u,
                                           w_i, bih_i, bhh_i,
                                           out);
}